// FusedMulSumCatReplacement_61933428408454
// MI455X (gfx1250) — compile-verified
//
#include <hip/hip_runtime.h>

// FusedMulSumCat: out[b, p*128 + n] = sum_k m{2p}[b,k,n] * m{2p+1}[b,k,n]
// B=4096, S1=64, S2=128, fp32 in / fp32 out.
//
// Memory-bound (AI ~0.26 F/B): stream 512MB once; reduce via
// V_WMMA_F32_16X16X4_F32 with an all-ones A matrix (4-row column reducer).

typedef __attribute__((ext_vector_type(2))) float v2f;
typedef __attribute__((ext_vector_type(8))) float v8f;

#define S1 64
#define S2 128

__global__ __launch_bounds__(256) void fused_mulsum_cat_wmma(
    const float* __restrict__ m0, const float* __restrict__ m1,
    const float* __restrict__ m2, const float* __restrict__ m3,
    float* __restrict__ out, int nB)
{
    const int gtid = blockIdx.x * blockDim.x + threadIdx.x;
    const int wave = gtid >> 5;              // global wave id (wave32)
    const int lane = threadIdx.x & 31;

    // wave-task decode: 8 column chunks x 2 pairs x nB batches
    const int colchunk = wave & 7;           // 16-column strip of the 128
    const int pair     = (wave >> 3) & 1;    // 0: m0*m1, 1: m2*m3
    const int b        = wave >> 4;
    if (b >= nB) return;                     // wave-uniform guard (no EXEC split)

    const float* __restrict__ X = pair ? m2 : m0;
    const float* __restrict__ Y = pair ? m3 : m1;
    const size_t tile = (size_t)b * (S1 * S2);

    const int n    = lane & 15;              // output column within strip (B-matrix N)
    const int koff = (lane >> 4) << 1;       // lanes 16-31 take K rows +2/+3
    const int col  = colchunk * 16 + n;

    const float* px = X + tile + (size_t)koff * S2 + col;
    const float* py = Y + tile + (size_t)koff * S2 + col;

    v2f ones;  ones.x = 1.0f;  ones.y = 1.0f;   // A matrix: 16x4 all-ones
    v8f acc = {};                               // C/D accumulator, 16x16 f32

#pragma unroll 4
    for (int k0 = 0; k0 < S1; k0 += 4) {
        // B matrix 4x16: this lane supplies 2 of the 4 K-rows for column `col`.
        // With A==1 the K<->slot permutation is irrelevant; only lane->N matters.
        v2f prod;
        prod.x = px[0]  * py[0];
        prod.y = px[S2] * py[S2];
        // D = ones(16x4) x prod(4x16) + C  => every row of D = column sums
        acc = __builtin_amdgcn_wmma_f32_16x16x4_f32(
            /*neg_a=*/false, ones, /*neg_b=*/false, prod,
            /*c_mod=*/(short)0, acc, /*reuse_a=*/false, /*reuse_b=*/false);
        px += 4 * S2;
        py += 4 * S2;
    }

    // D layout: VGPR0, lanes 0-15 hold M=0, N=lane -> the column sum.
    if (lane < 16) {
        out[(size_t)b * (2 * S2) + (size_t)pair * S2 + col] = acc[0];
    }
}

extern "C" void kernel_launch(void* const* d_in, const int* in_sizes, int n_in,
                              void* d_out, int out_size, void* d_ws, size_t ws_size,
                              hipStream_t stream) {
    const float* m0 = (const float*)d_in[0];
    const float* m1 = (const float*)d_in[1];
    const float* m2 = (const float*)d_in[2];
    const float* m3 = (const float*)d_in[3];
    float* out = (float*)d_out;

    const int nB = in_sizes[0] / (S1 * S2);        // 4096
    const int totalWaves = nB * 16;                // (8 chunks x 2 pairs) per b
    const int blocks = (totalWaves + 7) / 8;       // 8 waves per 256-thread block

    fused_mulsum_cat_wmma<<<blocks, 256, 0, stream>>>(m0, m1, m2, m3, out, nB);
}